// BipartiteGNNLayer_266287972797
// MI455X (gfx1250) — compile-verified
//
#include <hip/hip_runtime.h>
#include <hip/hip_bf16.h>

// ---------------------------------------------------------------------------
// Bipartite GraphConv layer for MI455X (gfx1250, wave32).
//   agg_var = segment_sum(x_con[e_con] -> e_var)        (atomic f32 scatter)
//   agg_con = segment_sum(x_var[e_var] -> e_con)
//   out_var = relu(LN(agg_var @ W_rel_cv + b_cv + x_var @ W_root_cv))
//   out_con = relu(LN(agg_con @ W_rel_vc + b_vc + x_con @ W_root_vc))
// GEMMs use V_WMMA_F32_16X16X4_F32; A-tiles staged to LDS via the Tensor
// Data Mover (TENSOR_LOAD_TO_LDS) with hardware row padding (stride 132 f32).
// ---------------------------------------------------------------------------

#define N_CON 100000
#define N_VAR 200000
#define NEDGE 1600000
#define DIM   128

typedef __attribute__((ext_vector_type(2))) float v2f;
typedef __attribute__((ext_vector_type(8))) float v8f;
typedef __attribute__((ext_vector_type(4))) unsigned int u32x4;
typedef __attribute__((ext_vector_type(8))) int i32x8;
typedef __attribute__((ext_vector_type(4))) int i32x4;

// ---------------- zero workspace ----------------
__global__ __launch_bounds__(256)
void zero_kernel(float4* __restrict__ p, long n4) {
    long t = (long)blockIdx.x * blockDim.x + threadIdx.x;
    if (t < n4) p[t] = make_float4(0.f, 0.f, 0.f, 0.f);
}

// ---------------- edge scatter-add ----------------
// one thread per (edge, 4-float chunk); 32 chunks cover D=128.
__global__ __launch_bounds__(256)
void scatter_add_kernel(const float* __restrict__ feats,
                        const int* __restrict__ src_idx,
                        const int* __restrict__ dst_idx,
                        float* __restrict__ agg) {
    long t = (long)blockIdx.x * 256 + threadIdx.x;
    if (t >= (long)NEDGE * 32) return;
    int e = (int)(t >> 5);
    int c = (int)(t & 31) << 2;              // float offset within row
    long soff = (long)src_idx[e] * DIM + c;
    long doff = (long)dst_idx[e] * DIM + c;
    float4 v = *(const float4*)(feats + soff);
    __hip_atomic_fetch_add(agg + doff + 0, v.x, __ATOMIC_RELAXED, __HIP_MEMORY_SCOPE_AGENT);
    __hip_atomic_fetch_add(agg + doff + 1, v.y, __ATOMIC_RELAXED, __HIP_MEMORY_SCOPE_AGENT);
    __hip_atomic_fetch_add(agg + doff + 2, v.z, __ATOMIC_RELAXED, __HIP_MEMORY_SCOPE_AGENT);
    __hip_atomic_fetch_add(agg + doff + 3, v.w, __ATOMIC_RELAXED, __HIP_MEMORY_SCOPE_AGENT);
}

// ---------------- TDM: load one 16x128 f32 tile into LDS ----------------
// D# per CDNA5 ISA ch.8. Row-major tile (tile_dim0=128 elems, tile_dim1=16),
// tensor_dim0_stride=128, data_size=4B. pad_enable with pad_interval=6
// (every 512B = one row) and pad_amount=3 (+4 DWORDs) -> LDS row stride of
// 132 floats, matching the bank-conflict-free layout used by the WMMA feed.
__device__ __forceinline__ void tdm_load_tile_16x128(const float* gptr,
                                                     unsigned lds_off) {
    unsigned long long ga = (unsigned long long)(uintptr_t)gptr;
    u32x4 g0;
    g0[0] = 1u;                                        // count=1, no gather
    g0[1] = lds_off;                                   // lds_addr [63:32]
    g0[2] = (unsigned)(ga & 0xFFFFFFFFull);            // global_addr[31:0]
    g0[3] = (unsigned)((ga >> 32) & 0x01FFFFFFull)     // global_addr[56:32]
            | (2u << 30);                              // type=2 ("image")
    i32x8 g1;
    g1[0] = (int)((2u << 16)        // data_size = 4 bytes
            | (1u << 20)            // pad_enable
            | (6u << 22)            // pad_interval: every 512 bytes
            | (3u << 25));          // pad_amount: 4 DWORDs
    g1[1] = (int)(128u << 16);      // tensor_dim0[15:0] = 128 (bits 63:48)
    g1[2] = (int)(16u << 16);       // tensor_dim1[15:0] = 16  (bits 95:80)
    g1[3] = (int)(128u << 16);      // tile_dim0 = 128         (bits 127:112)
    g1[4] = 16;                     // tile_dim1 = 16, tile_dim2 = 0
    g1[5] = 128;                    // tensor_dim0_stride low32 = 128
    g1[6] = 0;                      // stride0 hi / stride1 lo
    g1[7] = 0;                      // stride1 hi
    i32x4 z4 = {0, 0, 0, 0};
#if defined(__clang_major__) && (__clang_major__ >= 23)
    i32x8 z8 = {0, 0, 0, 0, 0, 0, 0, 0};
    __builtin_amdgcn_tensor_load_to_lds(g0, g1, z4, z4, z8, 0);
#else
    __builtin_amdgcn_tensor_load_to_lds(g0, g1, z4, z4, 0);
#endif
}

// ---------------- fused dual-GEMM + bias + LayerNorm + ReLU ----------------
// Block = 256 threads = 8 waves. Block owns 16 rows; wave w owns output
// columns [16w, 16w+16). C accumulates agg@Wrel + xroot@Wroot over K=128.
__global__ __launch_bounds__(256)
void gemm_ln_relu_kernel(const float* __restrict__ agg,
                         const float* __restrict__ xroot,
                         const float* __restrict__ Wrel,
                         const float* __restrict__ Wroot,
                         const float* __restrict__ bias,
                         const float* __restrict__ gamma,
                         const float* __restrict__ beta,
                         float* __restrict__ out) {
    __shared__ float sAagg[16 * 132];   // padded row stride (132 % 64 banks = 4)
    __shared__ float sAroot[16 * 132];
    __shared__ float sOut[16 * 128];
    __shared__ float sMean[16];
    __shared__ float sRstd[16];

    const int tid = threadIdx.x;
    const long rowbase = (long)blockIdx.x * 16;

    // Stage both 16x128 A-tiles into LDS via the Tensor Data Mover.
    // Wave 0 issues both DMAs (EXEC is ignored by TENSOR_LOAD_TO_LDS; the
    // branch keeps the other 7 waves from issuing duplicates), waits on
    // TENSORcnt, then the whole block syncs.
    if (tid < 32) {
        tdm_load_tile_16x128(agg   + rowbase * DIM,
                             (unsigned)(uintptr_t)&sAagg[0]);
        tdm_load_tile_16x128(xroot + rowbase * DIM,
                             (unsigned)(uintptr_t)&sAroot[0]);
        __builtin_amdgcn_s_wait_tensorcnt(0);
    }
    __syncthreads();

    const int lane = tid & 31;
    const int wave = tid >> 5;
    const int hi   = lane >> 4;          // half-wave selector
    const int nlo  = lane & 15;
    const int ncol = wave * 16 + nlo;    // this lane's output column

    // A fragment: lane (0-15) holds row=lane, K={k0,k0+1}; lane (16-31) holds
    // row=lane-16, K={k0+2,k0+3}  (ISA 32-bit A 16x4 layout).
    const float* arow_a = sAagg  + nlo * 132;
    const float* arow_x = sAroot + nlo * 132;

    v8f c = {0.f, 0.f, 0.f, 0.f, 0.f, 0.f, 0.f, 0.f};
    #pragma unroll 4
    for (int ks = 0; ks < 32; ++ks) {
        const int k0 = ks * 4 + hi * 2;
        v2f a, b;
        // agg @ W_rel
        a.x = arow_a[k0];
        a.y = arow_a[k0 + 1];
        b.x = Wrel[(long)k0 * DIM + ncol];
        b.y = Wrel[(long)(k0 + 1) * DIM + ncol];
        c = __builtin_amdgcn_wmma_f32_16x16x4_f32(false, a, false, b,
                                                  (short)0, c, false, false);
        // x @ W_root
        a.x = arow_x[k0];
        a.y = arow_x[k0 + 1];
        b.x = Wroot[(long)k0 * DIM + ncol];
        b.y = Wroot[(long)(k0 + 1) * DIM + ncol];
        c = __builtin_amdgcn_wmma_f32_16x16x4_f32(false, a, false, b,
                                                  (short)0, c, false, false);
    }

    // Epilogue part 1: bias add, stage C to LDS.
    // C layout: VGPR i -> row = i + 8*hi, col = ncol.
    const float bv = bias[ncol];
    #pragma unroll
    for (int i = 0; i < 8; ++i) {
        int r = i + hi * 8;
        sOut[r * 128 + ncol] = c[i] + bv;
    }
    __syncthreads();

    // Epilogue part 2: per-row mean / rstd (population variance, eps=1e-5).
    if (tid < 16) {
        float s = 0.f, ss = 0.f;
        #pragma unroll 8
        for (int j = 0; j < 128; ++j) {
            float v = sOut[tid * 128 + j];
            s  += v;
            ss += v * v;
        }
        float m   = s * (1.f / 128.f);
        float var = ss * (1.f / 128.f) - m * m;
        sMean[tid] = m;
        sRstd[tid] = rsqrtf(var + 1e-5f);
    }
    __syncthreads();

    // Epilogue part 3: normalize + affine + ReLU + store (8 cols per thread).
    const int r     = tid >> 4;
    const int cbase = (tid & 15) * 8;
    const float m  = sMean[r];
    const float rs = sRstd[r];
    #pragma unroll
    for (int j = 0; j < 8; ++j) {
        int col = cbase + j;
        float v = (sOut[r * 128 + col] - m) * rs * gamma[col] + beta[col];
        out[(rowbase + r) * DIM + col] = fmaxf(v, 0.f);
    }
}

// ---------------------------------------------------------------------------
extern "C" void kernel_launch(void* const* d_in, const int* in_sizes, int n_in,
                              void* d_out, int out_size, void* d_ws, size_t ws_size,
                              hipStream_t stream) {
    const float* x_con     = (const float*)d_in[0];
    const float* x_var     = (const float*)d_in[1];
    const int*   e_con     = (const int*)  d_in[2];
    const int*   e_var     = (const int*)  d_in[3];
    const float* W_rel_cv  = (const float*)d_in[4];
    const float* b_cv      = (const float*)d_in[5];
    const float* W_root_cv = (const float*)d_in[6];
    const float* W_rel_vc  = (const float*)d_in[7];
    const float* b_vc      = (const float*)d_in[8];
    const float* W_root_vc = (const float*)d_in[9];
    const float* ln_g_con  = (const float*)d_in[10];
    const float* ln_b_con  = (const float*)d_in[11];
    const float* ln_g_var  = (const float*)d_in[12];
    const float* ln_b_var  = (const float*)d_in[13];

    float* agg_var = (float*)d_ws;                       // [N_VAR, 128]
    float* agg_con = agg_var + (long)N_VAR * DIM;        // [N_CON, 128]

    float* out_con = (float*)d_out;                      // [N_CON, 128]
    float* out_var = out_con + (long)N_CON * DIM;        // [N_VAR, 128]

    // 1) zero aggregation buffers (ws is poisoned).
    {
        long n4 = ((long)(N_VAR + N_CON) * DIM) / 4;     // 9,600,000 float4s
        int blocks = (int)((n4 + 255) / 256);
        zero_kernel<<<blocks, 256, 0, stream>>>((float4*)d_ws, n4);
    }

    // 2) edge scatter-adds (both directions).
    {
        long nt = (long)NEDGE * 32;
        int blocks = (int)((nt + 255) / 256);
        // constraint -> variable: gather x_con[e_con], add into agg_var[e_var]
        scatter_add_kernel<<<blocks, 256, 0, stream>>>(x_con, e_con, e_var, agg_var);
        // variable -> constraint: gather x_var[e_var], add into agg_con[e_con]
        scatter_add_kernel<<<blocks, 256, 0, stream>>>(x_var, e_var, e_con, agg_con);
    }

    // 3) fused dual-GEMM + LayerNorm + ReLU per node type.
    gemm_ln_relu_kernel<<<N_VAR / 16, 256, 0, stream>>>(
        agg_var, x_var, W_rel_cv, W_root_cv, b_cv, ln_g_var, ln_b_var, out_var);
    gemm_ln_relu_kernel<<<N_CON / 16, 256, 0, stream>>>(
        agg_con, x_con, W_rel_vc, W_root_vc, b_vc, ln_g_con, ln_b_con, out_con);
}